// UGLR_22084721836177
// MI455X (gfx1250) — compile-verified
//
#include <hip/hip_runtime.h>
#include <stdint.h>

// Problem constants (fixed by the reference).
#define ITERS 10
#define Bn 4
#define Cn 32
#define Hn 256
#define Wn 256
#define PLANE (Hn * Wn)

// Tiling: 64x64 LDS region, 10-deep halo for 10 fused iterations -> 44x44 output tile.
#define NC   4      // channels per workgroup
#define S    64     // staged region side
#define HALO 10
#define TILE 44     // S - 2*HALO
#define NTIL 6      // ceil(256/44)
#define THREADS 512 // 16 wave32s; each thread owns 1 column x 8 rows
#define RPT  8      // rows per thread

// LDS: ONLY the premixed x-buffer crosses threads. Channel-interleaved [S*S][NC].
#define SMEM_FLOATS (S * S * NC)        // 16384 floats
#define SMEM_BYTES  (SMEM_FLOATS * 4)   // 64 KB -> multiple workgroups per 320KB WGP

// gfx1250 async global->LDS copy (ASYNCcnt-tracked, GVS addressing:
// saddr = 64-bit uniform base, vaddr = per-lane byte offset, vdst = LDS byte addr).
__device__ __forceinline__ void async_ld_f32(uint32_t lds_byte, const float* sbase, int goff_bytes) {
    asm volatile("global_load_async_to_lds_b32 %0, %1, %2"
                 :: "v"(lds_byte), "v"(goff_bytes), "s"(sbase)
                 : "memory");
}
__device__ __forceinline__ void async_wait_all() {
    asm volatile("s_wait_asynccnt 0" ::: "memory");
}

extern "C" __global__ void __launch_bounds__(THREADS)
svconv_fused(const float* __restrict__ x_init,
             const float* __restrict__ ab,
             const float* __restrict__ mu,
             float* __restrict__ out)
{
    extern __shared__ float smem[];   // XBI: [S*S][NC]

    const int tid = threadIdx.x;
    const int j   = tid & (S - 1);   // region column 0..63
    const int rg  = tid >> 6;        // row group 0..7
    const int r0  = rg * RPT;        // first of 8 contiguous rows per thread

    const int tX = blockIdx.x % NTIL;
    const int tY = blockIdx.x / NTIL;
    const int cb = blockIdx.y * NC;  // channel base
    const int b  = blockIdx.z;

    const int h0 = tY * TILE - HALO; // region origin in image coords (may be negative)
    const int w0 = tX * TILE - HALO;

    // LDS byte offset of dynamic shared base (flat-LDS low 32 bits == LDS offset).
    const uint32_t lds0 = (uint32_t)(uintptr_t)smem;

    const int  wi    = w0 + j;
    const bool colIn = (wi >= 0) && (wi < Wn);

    const float* abB = ab + (size_t)b * 41 * PLANE;
    const float* muB = mu + (size_t)b * PLANE;
    const float* xiB = x_init + ((size_t)b * Cn + cb) * PLANE;

    // ---- 1) Kick off async staging of x_init into the interleaved LDS buffer.
    //         The copy engine fills LDS while the VALU does softmax setup below.
    //         Out-of-image pixels are zero-filled (== the reference's zero pad).
    #pragma unroll
    for (int k = 0; k < RPT; ++k) {
        const int i  = r0 + k;
        const int hi = h0 + i;
        const int p  = i * S + j;
        if (colIn && hi >= 0 && hi < Hn) {
            const int goff = (hi * Wn + wi) * 4;
            #pragma unroll
            for (int c = 0; c < NC; ++c)
                async_ld_f32(lds0 + 4u * (4 * p + c), xiB + (size_t)c * PLANE, goff);
        } else {
            *(float4*)(smem + 4 * p) = make_float4(0.f, 0.f, 0.f, 0.f);
        }
    }

    // ---- 2) Per-pixel softmax weights + premix coefficients, entirely in VGPRs.
    //         (Pixel p's weights are consumed only by p's owner thread, so they
    //          never need LDS; zeroing them outside the image keeps the iterated
    //          zero padding exact.)
    float wr[9][RPT], av[RPT], bv[RPT];
    #pragma unroll
    for (int k = 0; k < RPT; ++k) {
        const int  i  = r0 + k;
        const int  hi = h0 + i;
        const bool in = colIn && hi >= 0 && hi < Hn;
        const int  gp = hi * Wn + wi;

        float v[9];
        #pragma unroll
        for (int t = 0; t < 9; ++t) v[t] = in ? abB[(size_t)t * PLANE + gp] : 0.0f;
        float m = v[0];
        #pragma unroll
        for (int t = 1; t < 9; ++t) m = fmaxf(m, v[t]);
        float ssum = 0.0f;
        #pragma unroll
        for (int t = 0; t < 9; ++t) { v[t] = __expf(v[t] - m); ssum += v[t]; }
        const float inv = in ? (1.0f / ssum) : 0.0f;
        #pragma unroll
        for (int t = 0; t < 9; ++t) wr[t][k] = v[t] * inv;

        const float mv = in ? muB[gp] : 0.0f;
        const float rr = 1.0f / (mv + 0.1f);
        av[k] = in ? (mv * rr) : 0.0f;       // a = mu/(mu+0.1)
        bv[k] = in ? (0.1f * rr) : 0.0f;     // b = 0.1/(mu+0.1)
    }

    // Prefetch bias planes into L2 while the 10 iterations run (one lane per 16 cols).
    if (colIn && j >= HALO && j < HALO + TILE && ((j - HALO) & 15) == 0) {
        #pragma unroll
        for (int c = 0; c < NC; ++c) {
            const float* bp = abB + (size_t)(9 + cb + c) * PLANE;
            #pragma unroll
            for (int k = 0; k < RPT; ++k) {
                const int i  = r0 + k;
                const int hi = h0 + i;
                if (i >= HALO && i < HALO + TILE && hi < Hn)
                    __builtin_prefetch(bp + (size_t)hi * Wn + wi, 0, 1);
            }
        }
    }

    // ---- 3) Wait for our own async copies, then pull iteration state into VGPRs.
    //         No barrier needed: each pixel is read/written only by its owner until
    //         the premix->conv handoff inside the loop.
    async_wait_all();
    float xr[NC][RPT], bxi[NC][RPT];
    #pragma unroll
    for (int k = 0; k < RPT; ++k) {
        const int p = (r0 + k) * S + j;
        const float4 x4 = *(const float4*)(smem + 4 * p);
        xr[0][k] = x4.x;  xr[1][k] = x4.y;  xr[2][k] = x4.z;  xr[3][k] = x4.w;
        bxi[0][k] = bv[k] * x4.x;  bxi[1][k] = bv[k] * x4.y;   // b*x_init, reused
        bxi[2][k] = bv[k] * x4.z;  bxi[3][k] = bv[k] * x4.w;   // every iteration
    }

    // ---- 4) 10 fused premix + spatially-varying 3x3 conv iterations ----
    for (int it = 0; it < ITERS; ++it) {
        // premix: single FMA per channel, one ds_store_b128 per pixel
        #pragma unroll
        for (int k = 0; k < RPT; ++k) {
            const int p = (r0 + k) * S + j;
            float4 px;
            px.x = fmaf(av[k], xr[0][k], bxi[0][k]);
            px.y = fmaf(av[k], xr[1][k], bxi[1][k]);
            px.z = fmaf(av[k], xr[2][k], bxi[2][k]);
            px.w = fmaf(av[k], xr[3][k], bxi[3][k]);
            *(float4*)(smem + 4 * p) = px;
        }
        __syncthreads();

        // conv: rolling 3-row window, weights from registers, each neighbor fetch is
        // one ds_load_b128 covering all 4 channels. Halo rows/cols produce garbage
        // that stays strictly inside the 10-deep discarded halo.
        float4 t0[3], t1[3];
        {
            const int q = (r0 - 1) * S + j;
            t0[0] = *(const float4*)(smem + 4 * (q - 1));
            t0[1] = *(const float4*)(smem + 4 * q);
            t0[2] = *(const float4*)(smem + 4 * (q + 1));
            t1[0] = *(const float4*)(smem + 4 * (q + S - 1));
            t1[1] = *(const float4*)(smem + 4 * (q + S));
            t1[2] = *(const float4*)(smem + 4 * (q + S + 1));
        }
        #pragma unroll
        for (int k = 0; k < RPT; ++k) {
            const int i  = r0 + k;
            const int qn = (i + 1) * S + j;
            const float4 n0 = *(const float4*)(smem + 4 * (qn - 1));
            const float4 n1 = *(const float4*)(smem + 4 * qn);
            const float4 n2 = *(const float4*)(smem + 4 * (qn + 1));

            xr[0][k] = wr[0][k] * t0[0].x + wr[1][k] * t0[1].x + wr[2][k] * t0[2].x
                     + wr[3][k] * t1[0].x + wr[4][k] * t1[1].x + wr[5][k] * t1[2].x
                     + wr[6][k] * n0.x    + wr[7][k] * n1.x    + wr[8][k] * n2.x;
            xr[1][k] = wr[0][k] * t0[0].y + wr[1][k] * t0[1].y + wr[2][k] * t0[2].y
                     + wr[3][k] * t1[0].y + wr[4][k] * t1[1].y + wr[5][k] * t1[2].y
                     + wr[6][k] * n0.y    + wr[7][k] * n1.y    + wr[8][k] * n2.y;
            xr[2][k] = wr[0][k] * t0[0].z + wr[1][k] * t0[1].z + wr[2][k] * t0[2].z
                     + wr[3][k] * t1[0].z + wr[4][k] * t1[1].z + wr[5][k] * t1[2].z
                     + wr[6][k] * n0.z    + wr[7][k] * n1.z    + wr[8][k] * n2.z;
            xr[3][k] = wr[0][k] * t0[0].w + wr[1][k] * t0[1].w + wr[2][k] * t0[2].w
                     + wr[3][k] * t1[0].w + wr[4][k] * t1[1].w + wr[5][k] * t1[2].w
                     + wr[6][k] * n0.w    + wr[7][k] * n1.w    + wr[8][k] * n2.w;

            t0[0] = t1[0]; t0[1] = t1[1]; t0[2] = t1[2];
            t1[0] = n0;    t1[1] = n1;    t1[2] = n2;
        }
        __syncthreads();
    }

    // ---- 5) Epilogue: add bias (already L2-hot from prefetch), store interior ----
    if (colIn && j >= HALO && j < HALO + TILE) {
        #pragma unroll
        for (int k = 0; k < RPT; ++k) {
            const int i  = r0 + k;
            const int hi = h0 + i;
            if (i >= HALO && i < HALO + TILE && hi < Hn) {
                const size_t pix = (size_t)hi * Wn + wi;
                #pragma unroll
                for (int c = 0; c < NC; ++c) {
                    const float bias = abB[(size_t)(9 + cb + c) * PLANE + pix];
                    out[((size_t)b * Cn + cb + c) * PLANE + pix] = xr[c][k] + bias;
                }
            }
        }
    }
}

extern "C" void kernel_launch(void* const* d_in, const int* in_sizes, int n_in,
                              void* d_out, int out_size, void* d_ws, size_t ws_size,
                              hipStream_t stream) {
    (void)in_sizes; (void)n_in; (void)d_ws; (void)ws_size; (void)out_size;
    const float* x_init = (const float*)d_in[0];
    const float* ab     = (const float*)d_in[1];
    const float* mu     = (const float*)d_in[2];
    // d_in[3] is K (==3), baked into the kernel.

    // 64KB dynamic LDS per workgroup. Idempotent, capture-safe.
    hipFuncSetAttribute(reinterpret_cast<const void*>(svconv_fused),
                        hipFuncAttributeMaxDynamicSharedMemorySize, SMEM_BYTES);

    dim3 grid(NTIL * NTIL, Cn / NC, Bn);   // 36 tiles x 8 channel groups x 4 batches
    svconv_fused<<<grid, THREADS, SMEM_BYTES, stream>>>(x_init, ab, mu, (float*)d_out);
}